// ExponentialSmoothingAttention_89318139887830
// MI455X (gfx1250) — compile-verified
//
#include <hip/hip_runtime.h>

// ---------------------------------------------------------------------------
// ExponentialSmoothingAttention on gfx1250 (MI455X)
//
// out[b,n,h*dH+c] = a*r^(N-1-n) * z[n] + r^(n+1)*v0[h,c]
//   z[n] = sum_{k<=n} x[b,k,h*dH+c]*r^k ,  a = sigmoid(alpha), r = 1-a
//
// Bandwidth-bound (~140-200 MB @ 23.3 TB/s => ~6-9 us). Prefix scan done as
// chunked scan; intra-chunk triangular prefix runs on the matrix pipe via
// V_WMMA_F32_16X16X4_F32 (scan-as-matmul with LowTri/Ones A fragments).
// exp2f minimized via shrink-only multiplicative ladders; all in-chunk
// addressing uses compile-time immediate offsets from one base pointer.
// ---------------------------------------------------------------------------

typedef float v2f __attribute__((ext_vector_type(2)));
typedef float v8f __attribute__((ext_vector_type(8)));

#define BB   8
#define NN   4096
#define DD   512
#define HH   8
#define dHH  64            // DD/HH
#define CHK  64            // time steps per chunk
#define NCHK (NN / CHK)    // 64 chunks

__device__ __forceinline__ float sigmoidf(float v) {
    return 1.0f / (1.0f + __expf(-v));
}

// --- Kernel 1: per-chunk sums S[bh][chunk][ch] = sum_j x*r^n ----------------
__global__ __launch_bounds__(256)
void k_chunk_sums(const float* __restrict__ x, const float* __restrict__ alpha,
                  float* __restrict__ S) {
    int blk = blockIdx.x;               // bh*NCHK + chunk
    int bh  = blk / NCHK, c = blk % NCHK;
    int b   = bh / HH,    h = bh % HH;
    int tid = threadIdx.x;
    int ch  = tid & 63;                 // channel within head
    int seg = tid >> 6;                 // 4 row segments of 16

    float a   = sigmoidf(alpha[0]);
    float r   = 1.0f - a;
    float l2r = __log2f(r);

    __shared__ float red[4][64];
    int n0 = c * CHK + seg * 16;
    // single base pointer; loop offsets are compile-time immediates
    const float* xq = x + (size_t)b * NN * DD + (size_t)n0 * DD
                        + (size_t)h * dHH + ch;
    float p = exp2f((float)n0 * l2r);   // r^n0 (ladder start)
    float s = 0.0f;
#pragma unroll
    for (int j = 0; j < 16; ++j) {
        s += xq[j * DD] * p;
        p *= r;                          // shrink-only ladder
    }
    red[seg][ch] = s;
    __syncthreads();
    if (seg == 0) {
        S[(size_t)blk * 64 + ch] = red[0][ch] + red[1][ch] + red[2][ch] + red[3][ch];
    }
}

// --- Kernel 2: exclusive prefix of S over chunks -> Cx ----------------------
__global__ __launch_bounds__(256)
void k_chunk_prefix(const float* __restrict__ S, float* __restrict__ Cx) {
    int t  = blockIdx.x * blockDim.x + threadIdx.x;   // 0..4095
    int bh = t >> 6, ch = t & 63;
    float carry = 0.0f;
    for (int c = 0; c < NCHK; ++c) {
        size_t idx = ((size_t)(bh * NCHK + c)) * 64 + ch;
        Cx[idx] = carry;
        carry  += S[idx];
    }
}

// --- Kernel 3: WMMA triangular prefix + output ------------------------------
// Block = 128 threads (4 waves). Wave w handles channels 16w..16w+15 over one
// 64-step chunk of one (b,h). A-fragment layout (16x4 f32): lane L -> M=L%16,
// VGPR0 holds K=2*(L/16), VGPR1 holds K=2*(L/16)+1. B (4x16): lane L -> N=L%16,
// VGPR0 row 2*(L/16), VGPR1 row 2*(L/16)+1. C/D: lane L col=L%16, VGPR v is
// row v + 8*(L/16).
__global__ __launch_bounds__(128)
void k_main(const float* __restrict__ x, const float* __restrict__ alpha,
            const float* __restrict__ v0, const float* __restrict__ Cx,
            float* __restrict__ out) {
    int blk  = blockIdx.x;              // bh*NCHK + chunk
    int bh   = blk / NCHK, c = blk % NCHK;
    int b    = bh / HH,    h = bh % HH;
    int lane = threadIdx.x & 31;
    int w    = threadIdx.x >> 5;        // wave -> channel group
    int h2   = lane >> 4;               // lane half
    int m    = lane & 15;               // A-row / D-column index
    int chcol = (w << 4) | m;           // channel within head

    float a   = sigmoidf(alpha[0]);
    float r   = 1.0f - a;
    float l2r = __log2f(r);
    int   n0  = c * CHK;

    // small powers of r by squaring (no exp)
    float r2  = r * r;
    float r4  = r2 * r2;
    float r8  = r4 * r4;
    float r12 = r8 * r4;
    float r16 = r8 * r8;

    // Base pointers with n0 and the h2-dependent row term folded in, so all
    // unrolled accesses use compile-time immediate offsets (<= ~123 KB).
    size_t chan = (size_t)b * NN * DD + (size_t)h * dHH + chcol;
    const float* xrow = x   + chan + (size_t)(n0 + 2 * h2) * DD;  // B-frag rows
    float*       orow = out + chan + (size_t)(n0 + 8 * h2) * DD;  // D rows

    float carry = Cx[((size_t)(bh * NCHK + c)) * 64 + chcol];
    float v0v   = v0[h * dHH + chcol];

    // ladders (both shrink-only => underflow-to-zero is correct)
    float pbase = exp2f((float)(n0 + 2 * h2) * l2r);            // r^(n0+2h2)
    float ibase = exp2f((float)(n0 + 1 + 8 * h2) * l2r) * v0v;  // r^(n0+1+8h2)*v0

    v8f accR = {};                      // Ones x (sum of previous 16-blocks)

#pragma unroll
    for (int rb = 0; rb < 4; ++rb) {
        // Load + scale B fragments for the 4 K-slices of this 16-row block.
        v2f Bf[4];
        float rs4[4] = {1.0f, r4, r8, r12};
#pragma unroll
        for (int s = 0; s < 4; ++s) {
            float x0 = xrow[(rb * 16 + s * 4) * DD];       // row n0+rb16+s4+2h2
            float x1 = xrow[(rb * 16 + s * 4 + 1) * DD];
            float ps = pbase * rs4[s];                     // r^row
            Bf[s].x = x0 * ps;
            Bf[s].y = x1 * (ps * r);
        }
        pbase *= r16;

        // Cacc = accR + LowTri16 * T[rb]   (inclusive local prefix)
        v8f Cacc = accR;
#pragma unroll
        for (int s = 0; s < 4; ++s) {
            int k0 = s * 4 + 2 * h2;    // global K of A VGPR0 element
            v2f A;
            A.x = (m >= k0)     ? 1.0f : 0.0f;
            A.y = (m >= k0 + 1) ? 1.0f : 0.0f;
            Cacc = __builtin_amdgcn_wmma_f32_16x16x4_f32(
                false, A, false, Bf[s], (short)0, Cacc, false, false);
        }

        // accR += Ones16 * T[rb]  (running column sums for later blocks)
        if (rb < 3) {
            v2f One; One.x = 1.0f; One.y = 1.0f;
#pragma unroll
            for (int s = 0; s < 4; ++s) {
                accR = __builtin_amdgcn_wmma_f32_16x16x4_f32(
                    false, One, false, Bf[s], (short)0, accR, false, false);
            }
        }

        // Row factors a*r^(N-1-n): grow with n, so ladder DOWNWARD only
        // within the block (shrink-only); fresh exp2f per 16-row block.
        float rowf[8];
        rowf[7] = a * exp2f((float)(NN - 1 - (n0 + rb * 16 + 7 + 8 * h2)) * l2r);
#pragma unroll
        for (int v = 6; v >= 0; --v) rowf[v] = rowf[v + 1] * r;

        // Emit 16 output rows: row m' = v + 8*h2, col = chcol.
        float iv = ibase;               // r^(nrow+1)*v0 at v=0 of this block
#pragma unroll
        for (int v = 0; v < 8; ++v) {
            float z = carry + Cacc[v];
            orow[(rb * 16 + v) * DD] = rowf[v] * z + iv;
            iv *= r;
        }
        ibase *= r16;
    }
}

// ---------------------------------------------------------------------------
extern "C" void kernel_launch(void* const* d_in, const int* in_sizes, int n_in,
                              void* d_out, int out_size, void* d_ws, size_t ws_size,
                              hipStream_t stream) {
    const float* x     = (const float*)d_in[0];   // (8,4096,512) f32
    const float* alpha = (const float*)d_in[1];   // scalar f32 (pre-sigmoid)
    const float* v0    = (const float*)d_in[2];   // (1,8,64) f32
    // d_in[3] = heads (int, ==8), shapes hardcoded to match.
    float* out = (float*)d_out;

    // ws layout: S (64 bh * 64 chunks * 64 ch floats = 1 MB) | Cx (1 MB)
    float* S  = (float*)d_ws;
    float* Cx = S + (size_t)BB * HH * NCHK * 64;

    (void)in_sizes; (void)n_in; (void)out_size; (void)ws_size;

    k_chunk_sums<<<dim3(BB * HH * NCHK), 256, 0, stream>>>(x, alpha, S);

    k_chunk_prefix<<<dim3((BB * HH * 64) / 256), 256, 0, stream>>>(S, Cx);

    k_main<<<dim3(BB * HH * NCHK), 128, 0, stream>>>(x, alpha, v0, Cx, out);
}